// Motion_loss_54623394071025
// MI455X (gfx1250) — compile-verified
//
#include <hip/hip_runtime.h>
#include <stdint.h>

// ---------------------------------------------------------------------------
// Motion FK loss for MI455X (gfx1250, wave32).
//  kernel 1: per-frame FK + loss partials, TDM-staged LDS tiles, block reduce
//  kernel 2: deterministic final reduction using V_WMMA_F32_16X16X4_F32
// ---------------------------------------------------------------------------

#define BLOCK 64            // threads per block = frames per block (2 waves)
#define ROW   99            // floats per frame (24*4 quat + 3 pos)
#define TILE_ELEMS (BLOCK * ROW)   // 6336 f32 per input tile

typedef __attribute__((ext_vector_type(2))) float        v2f;
typedef __attribute__((ext_vector_type(8))) float        v8f;
typedef __attribute__((ext_vector_type(4))) unsigned int u32x4;
typedef __attribute__((ext_vector_type(4))) int          i32x4;
typedef __attribute__((ext_vector_type(8))) int          i32x8;

#if defined(__has_builtin)
#  if __has_builtin(__builtin_amdgcn_tensor_load_to_lds) && \
      __has_builtin(__builtin_amdgcn_s_wait_tensorcnt)
#    define HAVE_TDM 1
#    if __has_include(<hip/amd_detail/amd_gfx1250_TDM.h>)
#      define TDM_SIX_ARGS 1      /* amdgpu-toolchain: 6-arg builtin */
#    endif
#  endif
#  if __has_builtin(__builtin_amdgcn_wmma_f32_16x16x4_f32)
#    define HAVE_WMMA_F32 1
#  endif
#endif

struct M3 { float a[9]; };
struct V3 { float v[3]; };
struct Sk { M3 R; V3 t; };

// --------------------------- TDM 1-D tile load ------------------------------
#if defined(HAVE_TDM)
// Copy `nelems` contiguous f32 from global `gptr` to LDS byte-offset
// `lds_byte_off` with one Tensor Data Mover op.  D# packing per CDNA5 ISA
// 8.3/8.4: group0 = {count=1, lds_addr, global_addr[56:0], type=2},
// group1 = {wg_mask=0, data_size=4B, dims: 1 x nelems tile of a 1 x nelems
// tensor, dim0_stride=nelems}.  Groups 2/3 unused (zero).
__device__ __forceinline__ void tdm_load_f32_1d(uint32_t lds_byte_off,
                                                const float* gptr,
                                                uint32_t nelems) {
  uint64_t ga = (uint64_t)(uintptr_t)gptr;
  u32x4 g0;
  g0[0] = 1u;                                              // count=1, user D#
  g0[1] = lds_byte_off;                                    // lds_addr [63:32]
  g0[2] = (uint32_t)ga;                                    // global_addr lo
  g0[3] = (uint32_t)((ga >> 32) & 0x01FFFFFFu) | 0x80000000u; // hi25 | type=2
  i32x8 g1;
  g1[0] = (int)(2u << 16);                                 // data_size = 4B
  g1[1] = (int)((nelems & 0xFFFFu) << 16);                 // tensor_dim0 lo16
  g1[2] = (int)(((nelems >> 16) & 0xFFFFu) | (1u << 16));  // dim0 hi16, dim1=1
  g1[3] = (int)((nelems & 0xFFFFu) << 16);                 // tile_dim0
  g1[4] = 1;                                               // tile_dim1 = 1
  g1[5] = (int)nelems;                                     // dim0_stride lo32
  g1[6] = 0;
  g1[7] = 0;
  i32x4 z4 = (i32x4)0;
#if defined(TDM_SIX_ARGS)
  i32x8 z8 = (i32x8)0;
  __builtin_amdgcn_tensor_load_to_lds(g0, g1, z4, z4, z8, 0);
#else
  __builtin_amdgcn_tensor_load_to_lds(g0, g1, z4, z4, 0);
#endif
}
#endif

// ------------------------------ math helpers --------------------------------
__device__ __forceinline__ M3 quat2mat_s(float w, float x, float y, float z,
                                         float s) {
  M3 m;
  m.a[0] = 1.0f - s * (y * y + z * z);
  m.a[1] = s * (x * y - z * w);
  m.a[2] = s * (x * z + y * w);
  m.a[3] = s * (x * y + z * w);
  m.a[4] = 1.0f - s * (x * x + z * z);
  m.a[5] = s * (y * z - x * w);
  m.a[6] = s * (x * z - y * w);
  m.a[7] = s * (y * z + x * w);
  m.a[8] = 1.0f - s * (x * x + y * y);
  return m;
}

// One rsqrt per quaternion: inv = 1/||q|| feeds both the rot-loss
// normalization and two_s = 2/||q||^2 = 2*inv*inv.
__device__ __forceinline__ void quat_pair(const float* fx, const float* fy,
                                          int j, M3& Lx, M3& Ly,
                                          float& s_rot) {
  float wx = fx[4 * j + 0], xx = fx[4 * j + 1];
  float yx = fx[4 * j + 2], zx = fx[4 * j + 3];
  float wy = fy[4 * j + 0], xy = fy[4 * j + 1];
  float yy = fy[4 * j + 2], zy = fy[4 * j + 3];
  float dx = wx * wx + xx * xx + yx * yx + zx * zx;
  float dy = wy * wy + xy * xy + yy * yy + zy * zy;
  float ix = rsqrtf(dx);
  float iy = rsqrtf(dy);
  float e0 = wy * iy - wx * ix;
  float e1 = xy * iy - xx * ix;
  float e2 = yy * iy - yx * ix;
  float e3 = zy * iy - zx * ix;
  s_rot += e0 * e0 + e1 * e1 + e2 * e2 + e3 * e3;
  Lx = quat2mat_s(wx, xx, yx, zx, 2.0f * ix * ix);
  Ly = quat2mat_s(wy, xy, yy, zy, 2.0f * iy * iy);
}

__device__ __forceinline__ M3 matmul3(const M3& A, const M3& B) {
  M3 C;
#pragma unroll
  for (int r = 0; r < 3; ++r) {
#pragma unroll
    for (int c = 0; c < 3; ++c) {
      C.a[3 * r + c] = A.a[3 * r + 0] * B.a[0 + c] +
                       A.a[3 * r + 1] * B.a[3 + c] +
                       A.a[3 * r + 2] * B.a[6 + c];
    }
  }
  return C;
}

__device__ __forceinline__ V3 matvec3(const M3& A, float x, float y, float z) {
  V3 o;
  o.v[0] = A.a[0] * x + A.a[1] * y + A.a[2] * z;
  o.v[1] = A.a[3] * x + A.a[4] * y + A.a[5] * z;
  o.v[2] = A.a[6] * x + A.a[7] * y + A.a[8] * z;
  return o;
}

// One FK step for joint j (j>=1) on both skeletons, in place.
// Translation uses the PARENT rotation (current sx.R/sy.R) before R update.
__device__ __forceinline__ void do_joint(int j, const float* fx,
                                         const float* fy, const float* xt,
                                         const float* yt, Sk& sx, Sk& sy,
                                         float& s_rot, float& s_gtr) {
  M3 Lx, Ly;
  quat_pair(fx, fy, j, Lx, Ly, s_rot);
  V3 ox = matvec3(sx.R, xt[3 * j + 0], xt[3 * j + 1], xt[3 * j + 2]);
  V3 oy = matvec3(sy.R, yt[3 * j + 0], yt[3 * j + 1], yt[3 * j + 2]);
  sx.t.v[0] += ox.v[0]; sx.t.v[1] += ox.v[1]; sx.t.v[2] += ox.v[2];
  sy.t.v[0] += oy.v[0]; sy.t.v[1] += oy.v[1]; sy.t.v[2] += oy.v[2];
  float g0 = sy.t.v[0] - sx.t.v[0];
  float g1 = sy.t.v[1] - sx.t.v[1];
  float g2 = sy.t.v[2] - sx.t.v[2];
  s_gtr += g0 * g0 + g1 * g1 + g2 * g2;
  sx.R = matmul3(sx.R, Lx);
  sy.R = matmul3(sy.R, Ly);
}

// ------------------------------- kernel 1 -----------------------------------
__global__ void __launch_bounds__(BLOCK) motion_loss_main(
    const float* __restrict__ Ym, const float* __restrict__ Xm,
    const float* __restrict__ Yt, const float* __restrict__ Xt,
    float* __restrict__ ws, int T, float c_rot, float c_gtr, float c_pos) {
  __shared__ __align__(16) float xm[TILE_ELEMS];
  __shared__ __align__(16) float ym[TILE_ELEMS];
  __shared__ float xt[72];
  __shared__ float yt[72];
  __shared__ float red[BLOCK / 32];

  const int tid = threadIdx.x;
  const long base = (long)blockIdx.x * BLOCK;   // first frame of this block
  const int b = (int)(base / T);                // batch index (uniform; T%BLOCK==0)
  const float* gx = Xm + base * ROW;
  const float* gy = Ym + base * ROW;

#if defined(HAVE_TDM)
  // Wave 0 DMAs the X tile, wave 1 the Y tile; each waits on its TENSORcnt.
  if ((tid >> 5) == 0) {
    tdm_load_f32_1d((uint32_t)(uintptr_t)xm, gx, TILE_ELEMS);
  } else {
    tdm_load_f32_1d((uint32_t)(uintptr_t)ym, gy, TILE_ELEMS);
  }
  __builtin_amdgcn_s_wait_tensorcnt(0);
#else
  for (int i = tid; i < TILE_ELEMS; i += BLOCK) {
    xm[i] = gx[i];
    ym[i] = gy[i];
  }
#endif
  for (int i = tid; i < 72; i += BLOCK) {
    xt[i] = Xt[(long)b * 72 + i];
    yt[i] = Yt[(long)b * 72 + i];
  }
  __syncthreads();

  // Per-lane frame rows in LDS; stride 99 dwords => conflict-free column reads.
  const float* fx = &xm[tid * ROW];
  const float* fy = &ym[tid * ROW];

  float s_rot = 0.f, s_gtr = 0.f, s_pos = 0.f;
  Sk sx, sy, r0x, r0y, r9x, r9y;

  // Root joint 0: grot = local rot, gtr = raw motion position (== pos term).
  {
    M3 Lx, Ly;
    quat_pair(fx, fy, 0, Lx, Ly, s_rot);
    r0x.R = Lx; r0y.R = Ly;
    r0x.t.v[0] = fx[96]; r0x.t.v[1] = fx[97]; r0x.t.v[2] = fx[98];
    r0y.t.v[0] = fy[96]; r0y.t.v[1] = fy[97]; r0y.t.v[2] = fy[98];
    float d0 = r0y.t.v[0] - r0x.t.v[0];
    float d1 = r0y.t.v[1] - r0x.t.v[1];
    float d2 = r0y.t.v[2] - r0x.t.v[2];
    float dd = d0 * d0 + d1 * d1 + d2 * d2;
    s_pos = dd;          // mean((pos_y-pos_x)^2) term
    s_gtr += dd;         // joint-0 contribution to gtr MSE
  }

  // Topology = three chains off joint 0, three chains off joint 9.
  sx = r0x; sy = r0y;
  {
    const int ch[4] = {1, 4, 7, 10};
#pragma unroll
    for (int k = 0; k < 4; ++k)
      do_joint(ch[k], fx, fy, xt, yt, sx, sy, s_rot, s_gtr);
  }
  sx = r0x; sy = r0y;
  {
    const int ch[4] = {2, 5, 8, 11};
#pragma unroll
    for (int k = 0; k < 4; ++k)
      do_joint(ch[k], fx, fy, xt, yt, sx, sy, s_rot, s_gtr);
  }
  sx = r0x; sy = r0y;
  {
    const int ch[3] = {3, 6, 9};
#pragma unroll
    for (int k = 0; k < 3; ++k)
      do_joint(ch[k], fx, fy, xt, yt, sx, sy, s_rot, s_gtr);
  }
  r9x = sx; r9y = sy;      // save joint-9 state for the three sub-chains
  {
    const int ch[2] = {12, 15};
#pragma unroll
    for (int k = 0; k < 2; ++k)
      do_joint(ch[k], fx, fy, xt, yt, sx, sy, s_rot, s_gtr);
  }
  sx = r9x; sy = r9y;
  {
    const int ch[5] = {13, 16, 18, 20, 22};
#pragma unroll
    for (int k = 0; k < 5; ++k)
      do_joint(ch[k], fx, fy, xt, yt, sx, sy, s_rot, s_gtr);
  }
  sx = r9x; sy = r9y;
  {
    const int ch[5] = {14, 17, 19, 21, 23};
#pragma unroll
    for (int k = 0; k < 5; ++k)
      do_joint(ch[k], fx, fy, xt, yt, sx, sy, s_rot, s_gtr);
  }

  // Pre-scaled per-frame partial, then deterministic block reduction.
  float p = s_rot * c_rot + s_gtr * c_gtr + s_pos * c_pos;
#pragma unroll
  for (int o = 16; o > 0; o >>= 1) p += __shfl_down(p, o, 32);
  if ((tid & 31) == 0) red[tid >> 5] = p;
  __syncthreads();
  if (tid == 0) ws[blockIdx.x] = red[0] + red[1];
}

// ------------------------------- kernel 2 -----------------------------------
// Single wave sums the per-block partials in fixed lane order, then finishes
// the 32->1 reduction with V_WMMA_F32_16X16X4_F32 (A=ones => every D row in
// VGPR0 holds the 16 column sums of B; summing d[0] over all 32 lanes counts
// every B element exactly twice).
__global__ void __launch_bounds__(32) motion_final(const float* __restrict__ ws,
                                                   float* __restrict__ out,
                                                   int nblocks) {
  const int lane = threadIdx.x;
  float acc = 0.f;
  for (int i = lane; i < nblocks; i += 32) acc += ws[i];

  float tot;
#if defined(HAVE_WMMA_F32)
  v2f a;  a[0] = 1.0f; a[1] = 1.0f;       // A = ones (layout independent)
  v2f bv; bv[0] = acc; bv[1] = 0.0f;      // B holds the 32 partials (+zeros)
  v8f c = {};
  v8f d = __builtin_amdgcn_wmma_f32_16x16x4_f32(
      /*neg_a=*/false, a, /*neg_b=*/false, bv,
      /*c_mod=*/(short)0, c, /*reuse_a=*/false, /*reuse_b=*/false);
  float t = d[0];
#pragma unroll
  for (int o = 16; o > 0; o >>= 1) t += __shfl_xor(t, o, 32);
  tot = 0.5f * t;                          // each element counted twice
#else
  float t = acc;
#pragma unroll
  for (int o = 16; o > 0; o >>= 1) t += __shfl_xor(t, o, 32);
  tot = t;
#endif
  if (lane == 0) out[0] = tot;
}

// ------------------------------ host launcher -------------------------------
extern "C" void kernel_launch(void* const* d_in, const int* in_sizes, int n_in,
                              void* d_out, int out_size, void* d_ws,
                              size_t ws_size, hipStream_t stream) {
  const float* Ym = (const float*)d_in[0];   // (bs, T, 99)
  const float* Xm = (const float*)d_in[1];   // (bs, T, 99)
  const float* Yt = (const float*)d_in[2];   // (bs, 24, 3)
  const float* Xt = (const float*)d_in[3];   // (bs, 24, 3)

  const int frames = in_sizes[0] / ROW;      // bs*T
  const int bs = in_sizes[2] / 72;
  const int T = frames / bs;
  const int nblocks = frames / BLOCK;        // T % BLOCK == 0 in this problem

  // Loss scaling: B1=B2=1, w=ones.
  const double df = (double)frames;
  const float c_rot = (float)(1.0 / (df * 96.0));        // mean over rot elems
  const float c_gtr = (float)(2.5 / (df * 72.0));        // 2.5 * mean over gtr
  const float c_pos = (float)(1.0 / (df * 3.0));         // mean over pos elems

  float* ws = (float*)d_ws;
  motion_loss_main<<<nblocks, BLOCK, 0, stream>>>(Ym, Xm, Yt, Xt, ws, T, c_rot,
                                                  c_gtr, c_pos);
  motion_final<<<1, 32, 0, stream>>>(ws, (float*)d_out, nblocks);
}